// PatchMatch_76106820485558
// MI455X (gfx1250) — compile-verified
//
#include <hip/hip_runtime.h>
#include <math.h>

#define HH 240
#define WW 320
#define HW (HH * WW)            // 76800
#define DD 8
#define GG 8
#define CC 32
#define VV 4
#define NN 9
#define DHW (DD * HW)           // 614400
#define EPSB 1e-5f

// workspace float offsets
#define OFF_ROT   0             // 4 * 9
#define OFF_TR    36            // 4 * 3
#define OFF_PWS0  48
#define OFF_PWB0  64
#define OFF_PWS1  80
#define OFF_PWB1  88
#define OFF_SNS0  96
#define OFF_SNB0  112
#define OFF_SNS1  128
#define OFF_SNB1  136
#define OFF_XVOL  256                      // G * DHW floats
#define OFF_YVOL  (OFF_XVOL + GG * DHW)    // DHW floats
#define OFF_WSUM  (OFF_YVOL + DHW)         // HW floats

typedef __attribute__((ext_vector_type(2))) float v2f;
typedef __attribute__((ext_vector_type(8))) float v8f;

// ---------------------------------------------------------------------------
// K0: projection setup (inv(ref_proj), per-view rot/trans) + BN folding
// ---------------------------------------------------------------------------
__global__ void k0_setup(const float* __restrict__ ref_proj,
                         const float* __restrict__ src_projs,
                         const float* __restrict__ pw_bn0,
                         const float* __restrict__ pw_bn1,
                         const float* __restrict__ sn_bn0,
                         const float* __restrict__ sn_bn1,
                         float* __restrict__ ws) {
  if (threadIdx.x != 0 || blockIdx.x != 0) return;
  // Gauss-Jordan inverse of 4x4 (K-style matrix: diagonal is f,f,1,1 -> no pivot)
  float m[4][8];
  for (int r = 0; r < 4; ++r)
    for (int c = 0; c < 4; ++c) {
      m[r][c] = ref_proj[r * 4 + c];
      m[r][4 + c] = (r == c) ? 1.f : 0.f;
    }
  for (int p = 0; p < 4; ++p) {
    float inv = 1.f / m[p][p];
    for (int c = 0; c < 8; ++c) m[p][c] *= inv;
    for (int r = 0; r < 4; ++r) {
      if (r == p) continue;
      float f = m[r][p];
      for (int c = 0; c < 8; ++c) m[r][c] -= f * m[p][c];
    }
  }
  float inv4[16];
  for (int r = 0; r < 4; ++r)
    for (int c = 0; c < 4; ++c) inv4[r * 4 + c] = m[r][4 + c];

  for (int v = 0; v < VV; ++v) {
    float P[4][4];
    for (int i = 0; i < 4; ++i)
      for (int j = 0; j < 4; ++j) {
        float s = 0.f;
        for (int k = 0; k < 4; ++k) s += src_projs[v * 16 + i * 4 + k] * inv4[k * 4 + j];
        P[i][j] = s;
      }
    for (int i = 0; i < 3; ++i)
      for (int j = 0; j < 3; ++j) ws[OFF_ROT + v * 9 + i * 3 + j] = P[i][j];
    for (int i = 0; i < 3; ++i) ws[OFF_TR + v * 3 + i] = P[i][3];
  }
  // BN folds: s = g / sqrt(v + eps); b' = b - m*s   (stacked [g;b;m;v])
  for (int c = 0; c < 16; ++c) {
    float s = pw_bn0[c] / sqrtf(pw_bn0[48 + c] + EPSB);
    ws[OFF_PWS0 + c] = s;
    ws[OFF_PWB0 + c] = pw_bn0[16 + c] - pw_bn0[32 + c] * s;
  }
  for (int c = 0; c < 8; ++c) {
    float s = pw_bn1[c] / sqrtf(pw_bn1[24 + c] + EPSB);
    ws[OFF_PWS1 + c] = s;
    ws[OFF_PWB1 + c] = pw_bn1[8 + c] - pw_bn1[16 + c] * s;
  }
  for (int c = 0; c < 16; ++c) {
    float s = sn_bn0[c] / sqrtf(sn_bn0[48 + c] + EPSB);
    ws[OFF_SNS0 + c] = s;
    ws[OFF_SNB0 + c] = sn_bn0[16 + c] - sn_bn0[32 + c] * s;
  }
  for (int c = 0; c < 8; ++c) {
    float s = sn_bn1[c] / sqrtf(sn_bn1[24 + c] + EPSB);
    ws[OFF_SNS1 + c] = s;
    ws[OFF_SNB1 + c] = sn_bn1[8 + c] - sn_bn1[16 + c] * s;
  }
}

// ---------------------------------------------------------------------------
// K1: per-view warp + group similarity + pixel-wise net + accumulation
// one thread per pixel; launched once per view (sequential on stream)
// ---------------------------------------------------------------------------
__global__ __launch_bounds__(256) void k1_view(
    const float* __restrict__ ref, const float* __restrict__ src_all,
    const float* __restrict__ depth,
    const float* __restrict__ pw_w0, const float* __restrict__ pw_w1,
    const float* __restrict__ pw_w2, const float* __restrict__ pw_b2,
    float* __restrict__ ws, float* __restrict__ dout, int v) {
  __shared__ float sW0[128], sW1[128];
  __shared__ float sS0[16], sB0[16], sS1[8], sB1[8], sW2[8];
  __shared__ float sRot[9], sTr[3], sB2[1];
  const int tid = threadIdx.x;
  if (tid < 128) { sW0[tid] = pw_w0[tid]; sW1[tid] = pw_w1[tid]; }
  if (tid < 16) { sS0[tid] = ws[OFF_PWS0 + tid]; sB0[tid] = ws[OFF_PWB0 + tid]; }
  if (tid < 8) { sS1[tid] = ws[OFF_PWS1 + tid]; sB1[tid] = ws[OFF_PWB1 + tid]; sW2[tid] = pw_w2[tid]; }
  if (tid < 9) sRot[tid] = ws[OFF_ROT + v * 9 + tid];
  if (tid < 3) sTr[tid] = ws[OFF_TR + v * 3 + tid];
  if (tid == 0) sB2[0] = pw_b2[0];
  __syncthreads();

  const int p = blockIdx.x * 256 + tid;
  if (p >= HW) return;
  const int h = p / WW, w = p % WW;

  float refc[CC];
#pragma unroll
  for (int c = 0; c < CC; ++c) refc[c] = ref[c * HW + p];

  const float fw = (float)w, fh = (float)h;
  const float rx = sRot[0] * fw + sRot[1] * fh + sRot[2];
  const float ry = sRot[3] * fw + sRot[4] * fh + sRot[5];
  const float rz = sRot[6] * fw + sRot[7] * fh + sRot[8];
  const float* __restrict__ srcv = src_all + (size_t)v * CC * HW;

  float simb[DD][GG];
  float vmax = -1e30f;

#pragma unroll
  for (int d = 0; d < DD; ++d) {
    const float dep = depth[d * HW + p];
    float px = rx * dep + sTr[0];
    float py = ry * dep + sTr[1];
    float pz = rz * dep + sTr[2];
    if (pz <= 0.001f) { px = (float)WW; py = (float)HH; pz = 1.f; }
    const float gx = px / pz / ((WW - 1) * 0.5f) - 1.f;
    const float gy = py / pz / ((HH - 1) * 0.5f) - 1.f;
    const float ix = (gx + 1.f) * 0.5f * (float)(WW - 1);   // align_corners=True
    const float iy = (gy + 1.f) * 0.5f * (float)(HH - 1);
    const float x0f = floorf(ix), y0f = floorf(iy);
    const float wx1 = ix - x0f, wy1 = iy - y0f;
    const float x1f = x0f + 1.f, y1f = y0f + 1.f;
    const float vx0 = (x0f >= 0.f && x0f <= (float)(WW - 1)) ? 1.f : 0.f;
    const float vx1 = (x1f >= 0.f && x1f <= (float)(WW - 1)) ? 1.f : 0.f;
    const float vy0 = (y0f >= 0.f && y0f <= (float)(HH - 1)) ? 1.f : 0.f;
    const float vy1 = (y1f >= 0.f && y1f <= (float)(HH - 1)) ? 1.f : 0.f;
    const float w00 = (1.f - wx1) * (1.f - wy1) * vx0 * vy0;
    const float w10 = wx1 * (1.f - wy1) * vx1 * vy0;
    const float w01 = (1.f - wx1) * wy1 * vx0 * vy1;
    const float w11 = wx1 * wy1 * vx1 * vy1;
    const int xi0 = (int)fminf(fmaxf(x0f, 0.f), (float)(WW - 1));
    const int xi1 = (int)fminf(fmaxf(x1f, 0.f), (float)(WW - 1));
    const int yi0 = (int)fminf(fmaxf(y0f, 0.f), (float)(HH - 1));
    const int yi1 = (int)fminf(fmaxf(y1f, 0.f), (float)(HH - 1));
    const int o00 = yi0 * WW + xi0, o10 = yi0 * WW + xi1;
    const int o01 = yi1 * WW + xi0, o11 = yi1 * WW + xi1;

    float sim[GG];
#pragma unroll
    for (int g = 0; g < GG; ++g) sim[g] = 0.f;
#pragma unroll
    for (int c = 0; c < CC; ++c) {
      const float* sc = srcv + c * HW;
      const float wv = w00 * sc[o00] + w10 * sc[o10] + w01 * sc[o01] + w11 * sc[o11];
      sim[c >> 2] += wv * refc[c];
    }
#pragma unroll
    for (int g = 0; g < GG; ++g) { sim[g] *= 0.25f; simb[d][g] = sim[g]; }

    // pixel-wise net: 8 -> 16 -> 8 -> 1 (sigmoid)
    float h1[16];
#pragma unroll
    for (int j = 0; j < 16; ++j) {
      float acc = 0.f;
#pragma unroll
      for (int k = 0; k < 8; ++k) acc += sW0[j * 8 + k] * sim[k];
      h1[j] = fmaxf(acc * sS0[j] + sB0[j], 0.f);
    }
    float o = sB2[0];
#pragma unroll
    for (int j = 0; j < 8; ++j) {
      float acc = 0.f;
#pragma unroll
      for (int k = 0; k < 16; ++k) acc += sW1[j * 16 + k] * h1[k];
      const float h2 = fmaxf(acc * sS1[j] + sB1[j], 0.f);
      o += sW2[j] * h2;
    }
    const float sig = 1.f / (1.f + expf(-o));
    vmax = fmaxf(vmax, sig);
  }

  // view weight -> d_out (depth | score | view_weights)
  dout[HW + DHW + v * HW + p] = vmax;

  float* __restrict__ xvol = ws + OFF_XVOL;
  float* __restrict__ wsum = ws + OFF_WSUM;
  if (v == 0) {
#pragma unroll
    for (int d = 0; d < DD; ++d)
#pragma unroll
      for (int g = 0; g < GG; ++g)
        xvol[g * DHW + d * HW + p] = simb[d][g] * vmax;
    wsum[p] = vmax;
  } else {
#pragma unroll
    for (int d = 0; d < DD; ++d)
#pragma unroll
      for (int g = 0; g < GG; ++g)
        xvol[g * DHW + d * HW + p] += simb[d][g] * vmax;
    wsum[p] += vmax;
  }
}

// ---------------------------------------------------------------------------
// K2: similarity net (8 -> 16 -> 8 -> 1) via V_WMMA_F32_16X16X4_F32
// one wave handles 16 columns of X(8 x D*HW); all weights staged in LDS so
// the WMMA region has no divergent global loads / EXEC manipulation.
// ---------------------------------------------------------------------------
__global__ __launch_bounds__(256) void k2_simnet(
    const float* __restrict__ sn_w0, const float* __restrict__ sn_w1,
    const float* __restrict__ sn_w2, const float* __restrict__ sn_b2,
    float* __restrict__ ws) {
  __shared__ float sW0[128];    // W0 16x8
  __shared__ float sW1p[256];   // W1 zero-padded to 16x16 (rows 8..15 = 0)
  __shared__ float sS0[16], sB0[16], sS1[8], sB1[8], sW2[8], sB2[1];
  const int tid = threadIdx.x;
  if (tid < 128) sW0[tid] = sn_w0[tid];
  sW1p[tid] = (tid < 128) ? sn_w1[tid] : 0.f;
  if (tid < 16) { sS0[tid] = ws[OFF_SNS0 + tid]; sB0[tid] = ws[OFF_SNB0 + tid]; }
  if (tid < 8) { sS1[tid] = ws[OFF_SNS1 + tid]; sB1[tid] = ws[OFF_SNB1 + tid]; sW2[tid] = sn_w2[tid]; }
  if (tid == 0) sB2[0] = sn_b2[0];
  __syncthreads();

  const int lane = tid & 31;
  const int wid = tid >> 5;
  const int col0 = (blockIdx.x * 8 + wid) * 16;
  const float* __restrict__ xvol = ws + OFF_XVOL;
  const float* __restrict__ wsum = ws + OFF_WSUM;
  float* __restrict__ yvol = ws + OFF_YVOL;

  const bool lo = lane < 16;
  const int m = lane & 15;
  const int xcol = col0 + m;
  const float winv = 1.f / wsum[xcol % HW];  // sim_sum / w_sum, per pixel

  // ---- Layer 1: W0(16x8) x X(8x16). A layout 16x4: V0={K0|K2}, V1={K1|K3}
  v2f a0, a1, b0, b1;
  a0.x = sW0[m * 8 + (lo ? 0 : 2)];
  a0.y = sW0[m * 8 + (lo ? 1 : 3)];
  a1.x = sW0[m * 8 + 4 + (lo ? 0 : 2)];
  a1.y = sW0[m * 8 + 4 + (lo ? 1 : 3)];
  // B layout 4x16: V0={K0 lanes0-15 | K2 lanes16-31}, V1={K1|K3}
  b0.x = xvol[(size_t)(lo ? 0 : 2) * DHW + xcol] * winv;
  b0.y = xvol[(size_t)(lo ? 1 : 3) * DHW + xcol] * winv;
  b1.x = xvol[(size_t)(lo ? 4 : 6) * DHW + xcol] * winv;
  b1.y = xvol[(size_t)(lo ? 5 : 7) * DHW + xcol] * winv;

  v8f c1 = {0.f, 0.f, 0.f, 0.f, 0.f, 0.f, 0.f, 0.f};
  c1 = __builtin_amdgcn_wmma_f32_16x16x4_f32(false, a0, false, b0, (short)0, c1, false, false);
  c1 = __builtin_amdgcn_wmma_f32_16x16x4_f32(false, a1, false, b1, (short)0, c1, false, false);

  // BN + ReLU in C layout: VGPR j holds channel M = j + (lane<16 ? 0 : 8)
  const int hi8 = lo ? 0 : 8;
  float hv[8];
#pragma unroll
  for (int j = 0; j < 8; ++j) {
    hv[j] = fmaxf(c1[j] * sS0[j + hi8] + sB0[j + hi8], 0.f);
  }

  // Cross-half shuffles to convert C layout -> B layout for layer 2
  const int idxLo = lane & 15;        // lanes>=16 read lane-16
  const int idxHi = lane | 16;        // lanes<16 read lane+16
  const float sh2 = __shfl(hv[2], idxLo, 32);
  const float sh3 = __shfl(hv[3], idxLo, 32);
  const float sh6 = __shfl(hv[6], idxLo, 32);
  const float sh7 = __shfl(hv[7], idxLo, 32);
  const float sh0 = __shfl(hv[0], idxHi, 32);
  const float sh1 = __shfl(hv[1], idxHi, 32);
  const float sh4 = __shfl(hv[4], idxHi, 32);
  const float sh5 = __shfl(hv[5], idxHi, 32);

  // ---- Layer 2: W1(8x16, padded to 16 rows) x Y1(16x16); K = 16 -> 4 WMMAs
  v8f c2 = {0.f, 0.f, 0.f, 0.f, 0.f, 0.f, 0.f, 0.f};
#pragma unroll
  for (int kk = 0; kk < 4; ++kk) {
    v2f a, b;
    const int k0 = 4 * kk + (lo ? 0 : 2);
    const int k1 = 4 * kk + (lo ? 1 : 3);
    a.x = sW1p[m * 16 + k0];   // zero-padded rows: no divergence
    a.y = sW1p[m * 16 + k1];
    float bx, by;
    if (kk == 0)      { bx = lo ? hv[0] : sh2; by = lo ? hv[1] : sh3; }
    else if (kk == 1) { bx = lo ? hv[4] : sh6; by = lo ? hv[5] : sh7; }
    else if (kk == 2) { bx = lo ? sh0 : hv[2]; by = lo ? sh1 : hv[3]; }
    else              { bx = lo ? sh4 : hv[6]; by = lo ? sh5 : hv[7]; }
    b.x = bx; b.y = by;
    c2 = __builtin_amdgcn_wmma_f32_16x16x4_f32(false, a, false, b, (short)0, c2, false, false);
  }

  // ---- Layer 3: per pixel (lanes 0..15 hold channels M=0..7 of their column)
  if (lo) {
    float o = sB2[0];
#pragma unroll
    for (int j = 0; j < 8; ++j) {
      o += sW2[j] * fmaxf(c2[j] * sS1[j] + sB1[j], 0.f);
    }
    yvol[col0 + lane] = o;
  }
}

// ---------------------------------------------------------------------------
// K3: 9-offset bilinear resample of y, weighted sum over N, softmax over D,
// expected depth. One thread per pixel.
// ---------------------------------------------------------------------------
__global__ __launch_bounds__(256) void k3_score(
    const float* __restrict__ depth_sample, const float* __restrict__ grid,
    const float* __restrict__ weight, const float* __restrict__ ws,
    float* __restrict__ dout) {
  const int p = blockIdx.x * 256 + threadIdx.x;
  if (p >= HW) return;
  const int h = p / WW, w = p % WW;
  const float* __restrict__ yvol = ws + OFF_YVOL;

  float sv[DD];
#pragma unroll
  for (int d = 0; d < DD; ++d) sv[d] = 0.f;

  for (int n = 0; n < NN; ++n) {
    const int gbase = (((n * HH + h) * WW) + w) * 2;
    const float gx = grid[gbase + 0];
    const float gy = grid[gbase + 1];
    const float ix = ((gx + 1.f) * (float)WW - 1.f) * 0.5f;   // align_corners=False
    const float iy = ((gy + 1.f) * (float)HH - 1.f) * 0.5f;
    const float x0f = floorf(ix), y0f = floorf(iy);
    const float wx1 = ix - x0f, wy1 = iy - y0f;
    const int xi0 = (int)fminf(fmaxf(x0f, 0.f), (float)(WW - 1));
    const int xi1 = (int)fminf(fmaxf(x0f + 1.f, 0.f), (float)(WW - 1));
    const int yi0 = (int)fminf(fmaxf(y0f, 0.f), (float)(HH - 1));
    const int yi1 = (int)fminf(fmaxf(y0f + 1.f, 0.f), (float)(HH - 1));
    const float w00 = (1.f - wx1) * (1.f - wy1);
    const float w10 = wx1 * (1.f - wy1);
    const float w01 = (1.f - wx1) * wy1;
    const float w11 = wx1 * wy1;
    const int o00 = yi0 * WW + xi0, o10 = yi0 * WW + xi1;
    const int o01 = yi1 * WW + xi0, o11 = yi1 * WW + xi1;
#pragma unroll
    for (int d = 0; d < DD; ++d) {
      const float* yd = yvol + d * HW;
      const float s = w00 * yd[o00] + w10 * yd[o10] + w01 * yd[o01] + w11 * yd[o11];
      sv[d] += s * weight[(d * NN + n) * HW + p];
    }
  }

  float mx = sv[0];
#pragma unroll
  for (int d = 1; d < DD; ++d) mx = fmaxf(mx, sv[d]);
  float es[DD], sum = 0.f;
#pragma unroll
  for (int d = 0; d < DD; ++d) { es[d] = expf(sv[d] - mx); sum += es[d]; }
  const float inv = 1.f / sum;
  float dep = 0.f;
#pragma unroll
  for (int d = 0; d < DD; ++d) {
    const float sc = es[d] * inv;
    dout[HW + d * HW + p] = sc;
    dep += depth_sample[d * HW + p] * sc;
  }
  dout[p] = dep;
}

// ---------------------------------------------------------------------------
extern "C" void kernel_launch(void* const* d_in, const int* in_sizes, int n_in,
                              void* d_out, int out_size, void* d_ws, size_t ws_size,
                              hipStream_t stream) {
  (void)in_sizes; (void)n_in; (void)out_size; (void)ws_size;
  const float* ref_feature  = (const float*)d_in[0];
  const float* src_features = (const float*)d_in[1];
  const float* ref_proj     = (const float*)d_in[2];
  const float* src_projs    = (const float*)d_in[3];
  const float* depth_sample = (const float*)d_in[4];
  const float* grid         = (const float*)d_in[5];
  const float* weight       = (const float*)d_in[6];
  const float* pw_w0  = (const float*)d_in[7];
  const float* pw_bn0 = (const float*)d_in[8];
  const float* pw_w1  = (const float*)d_in[9];
  const float* pw_bn1 = (const float*)d_in[10];
  const float* pw_w2  = (const float*)d_in[11];
  const float* pw_b2  = (const float*)d_in[12];
  const float* sn_w0  = (const float*)d_in[13];
  const float* sn_bn0 = (const float*)d_in[14];
  const float* sn_w1  = (const float*)d_in[15];
  const float* sn_bn1 = (const float*)d_in[16];
  const float* sn_w2  = (const float*)d_in[17];
  const float* sn_b2  = (const float*)d_in[18];
  float* out = (float*)d_out;
  float* ws = (float*)d_ws;

  k0_setup<<<dim3(1), dim3(64), 0, stream>>>(ref_proj, src_projs, pw_bn0, pw_bn1,
                                             sn_bn0, sn_bn1, ws);
  for (int v = 0; v < VV; ++v) {
    k1_view<<<dim3(HW / 256), dim3(256), 0, stream>>>(
        ref_feature, src_features, depth_sample, pw_w0, pw_w1, pw_w2, pw_b2,
        ws, out, v);
  }
  k2_simnet<<<dim3(DHW / 128), dim3(256), 0, stream>>>(sn_w0, sn_w1, sn_w2, sn_b2, ws);
  k3_score<<<dim3(HW / 256), dim3(256), 0, stream>>>(depth_sample, grid, weight, ws, out);
}